// DensityAwareChamferLoss_12524124635648
// MI455X (gfx1250) — compile-verified
//
#include <hip/hip_runtime.h>
#include <hip/hip_bf16.h>
#include <math.h>

// ---------------- problem constants (match reference setup_inputs) ----------
constexpr int BATCH = 4;
constexpr int NPTS  = 8192;
constexpr float ALPHA_C = 1.0f;   // ALPHA
constexpr float EPS_C   = 1e-6f;  // EPS
constexpr float FRAC_C  = 1.0f;   // n_x/n_gt == n_gt/n_x == 1

// ---------------- kernel geometry -------------------------------------------
constexpr int THREADS       = 256;          // 8 wave32 per WG
constexpr int WAVES         = THREADS / 32;
constexpr int ROWS_PER_WAVE = 32;           // two 16-row A tiles per wave
constexpr int ROWS_PER_WG   = ROWS_PER_WAVE * WAVES;   // 256
constexpr int QCHUNK        = 2048;         // Q points staged in LDS (32 KB)

typedef float v2f __attribute__((ext_vector_type(2)));
typedef float v8f __attribute__((ext_vector_type(8)));

__device__ __forceinline__ void upd_min(float& mv, int& mi, float v, int j)
{
    if (v < mv) { mv = v; mi = j; }
}

// =============================================================================
// Kernel 1: for each P row, argmin_j ||P_i - Q_j||^2 via V_WMMA_F32_16X16X4_F32
//   tile math: D[i][j] = -2 P_i . Q_j + |Q_j|^2   (row-constant |P_i|^2 dropped)
//   Each wave: 32 rows (A0 rows 0-15, A1 rows 16-31); inner loop unrolled x2
//   columns -> 2 ds_load_b64 + 4 independent WMMAs per trip.
// =============================================================================
__global__ __launch_bounds__(THREADS)
void dac_argmin_kernel(const float* __restrict__ P, const float* __restrict__ Q,
                       int* __restrict__ outIdx, float* __restrict__ outDist)
{
    __shared__ float sq[QCHUNK * 4];   // per point: {-2x, -2y, -2z, |q|^2}

    const int wgPerBatch = NPTS / ROWS_PER_WG;          // 32
    const int b       = blockIdx.x / wgPerBatch;
    const int rowBase = (blockIdx.x % wgPerBatch) * ROWS_PER_WG;
    const int tid  = threadIdx.x;
    const int lane = tid & 31;
    const int wave = tid >> 5;
    const int half = lane >> 4;        // 0: lanes 0-15, 1: lanes 16-31
    const int l16  = lane & 15;

    const float* __restrict__ Pb = P + (size_t)b * NPTS * 3;
    const float* __restrict__ Qb = Q + (size_t)b * NPTS * 3;

    // --- A operands: row M = l%16, K = vgpr + 2*half ; rows [px,py,pz,1] ----
    const int waveRow = rowBase + wave * ROWS_PER_WAVE;
    const int rowA0 = waveRow + l16;
    const int rowA1 = waveRow + 16 + l16;
    const float p0x = Pb[rowA0 * 3 + 0], p0y = Pb[rowA0 * 3 + 1], p0z = Pb[rowA0 * 3 + 2];
    const float p1x = Pb[rowA1 * 3 + 0], p1y = Pb[rowA1 * 3 + 1], p1z = Pb[rowA1 * 3 + 2];
    v2f a0, a1;
    a0.x = half ? p0z  : p0x;          // K = 0 or 2
    a0.y = half ? 1.0f : p0y;          // K = 1 or 3
    a1.x = half ? p1z  : p1x;
    a1.y = half ? 1.0f : p1y;

    // min state: [0..7] = A0-block rows r+8*half, [8..15] = A1-block rows
    float minv[16];
    int   mini[16];
#pragma unroll
    for (int r = 0; r < 16; ++r) { minv[r] = 3.4e38f; mini[r] = 0; }

    for (int q0 = 0; q0 < NPTS; q0 += QCHUNK) {
        __syncthreads();
        // ---- stage Q chunk into LDS as {-2x,-2y,-2z,|q|^2} -----------------
        for (int j = tid; j < QCHUNK; j += THREADS) {
            const float qx = Qb[(q0 + j) * 3 + 0];
            const float qy = Qb[(q0 + j) * 3 + 1];
            const float qz = Qb[(q0 + j) * 3 + 2];
            sq[j * 4 + 0] = -2.0f * qx;
            sq[j * 4 + 1] = -2.0f * qy;
            sq[j * 4 + 2] = -2.0f * qz;
            sq[j * 4 + 3] = qx * qx + qy * qy + qz * qz;
        }
        if (q0 + QCHUNK < NPTS)        // hint next chunk -> global_prefetch_b8
            __builtin_prefetch(&Qb[(size_t)(q0 + QCHUNK) * 3], 0, 3);
        __syncthreads();

        // ---- sweep 16-column tiles, 2 tiles per trip -----------------------
        for (int jc = 0; jc < QCHUNK / 16; jc += 2) {
            const int jl0 = jc * 16 + l16;                // LDS-local columns
            const int jl1 = jl0 + 16;
            // B operand: column N = l%16, K = vgpr + 2*half (contiguous pair)
            const float2 bb0 = *(const float2*)&sq[jl0 * 4 + 2 * half];
            const float2 bb1 = *(const float2*)&sq[jl1 * 4 + 2 * half];
            v2f bv0; bv0.x = bb0.x; bv0.y = bb0.y;
            v2f bv1; bv1.x = bb1.x; bv1.y = bb1.y;

            v8f z = {0.f, 0.f, 0.f, 0.f, 0.f, 0.f, 0.f, 0.f};
            v8f acc00 = __builtin_amdgcn_wmma_f32_16x16x4_f32(
                false, a0, false, bv0, (short)0, z, false, false);
            v8f acc10 = __builtin_amdgcn_wmma_f32_16x16x4_f32(
                false, a1, false, bv0, (short)0, z, false, false);
            v8f acc01 = __builtin_amdgcn_wmma_f32_16x16x4_f32(
                false, a0, false, bv1, (short)0, z, false, false);
            v8f acc11 = __builtin_amdgcn_wmma_f32_16x16x4_f32(
                false, a1, false, bv1, (short)0, z, false, false);

            const int gj0 = q0 + jl0;                     // global column idx
            const int gj1 = gj0 + 16;
#pragma unroll
            for (int r = 0; r < 8; ++r) {
                upd_min(minv[r],     mini[r],     acc00[r], gj0);  // keeps first
                upd_min(minv[r],     mini[r],     acc01[r], gj1);  // (gj0<gj1)
                upd_min(minv[8 + r], mini[8 + r], acc10[r], gj0);
                upd_min(minv[8 + r], mini[8 + r], acc11[r], gj1);
            }
        }
    }

    // ---- cross-lane argmin over the 16-lane half (ties: smallest index) ----
#pragma unroll
    for (int r = 0; r < 16; ++r) {
        float v = minv[r]; int i = mini[r];
#pragma unroll
        for (int s = 1; s < 16; s <<= 1) {
            const float ov = __shfl_xor(v, s, 32);
            const int   oi = __shfl_xor(i, s, 32);
            if (ov < v || (ov == v && oi < i)) { v = ov; i = oi; }
        }
        minv[r] = v; mini[r] = i;
    }

    // ---- lane l16==0 of each half writes 16 rows; recompute exact distance -
    if (l16 == 0) {
#pragma unroll
        for (int r = 0; r < 16; ++r) {
            const int blk = r >> 3;                       // 0: A0 rows, 1: A1
            const int row = waveRow + blk * 16 + (r & 7) + 8 * half;
            const int j   = mini[r];
            const float dx = Pb[row * 3 + 0] - Qb[j * 3 + 0];
            const float dy = Pb[row * 3 + 1] - Qb[j * 3 + 1];
            const float dz = Pb[row * 3 + 2] - Qb[j * 3 + 2];
            outIdx [(size_t)b * NPTS + row] = j;
            outDist[(size_t)b * NPTS + row] = dx * dx + dy * dy + dz * dz;
        }
    }
}

// =============================================================================
// Kernel 2: zero the count buffers (harness poisons ws; must re-zero per call)
// =============================================================================
__global__ void dac_zero_kernel(unsigned int* __restrict__ buf, int n)
{
    const int t = blockIdx.x * blockDim.x + threadIdx.x;
    if (t < n) buf[t] = 0u;
}

// =============================================================================
// Kernel 3: density counts (integer atomics -> order-independent result)
// =============================================================================
__global__ void dac_count_kernel(const int* __restrict__ idx,
                                 unsigned int* __restrict__ count)
{
    const int t = blockIdx.x * blockDim.x + threadIdx.x;   // over BATCH*NPTS
    if (t < BATCH * NPTS) {
        const int b = t / NPTS;
        atomicAdd(&count[(size_t)b * NPTS + idx[t]], 1u);
    }
}

// =============================================================================
// Kernel 4: per-point loss term + deterministic block tree-sum
//   term = 1 - exp(-dist*ALPHA) * frac / (count[idx]^1 + eps)
// =============================================================================
__global__ __launch_bounds__(256)
void dac_loss_partial_kernel(const int* __restrict__ idx,
                             const float* __restrict__ dist,
                             const unsigned int* __restrict__ count,
                             float* __restrict__ partial)
{
    __shared__ float red[256];
    const int t = blockIdx.x * 256 + threadIdx.x;          // over BATCH*NPTS
    const int b = t / NPTS;
    const float c = (float)count[(size_t)b * NPTS + idx[t]];  // ^N_LAMBDA(=1)
    const float w = FRAC_C / (c + EPS_C);
    const float term = 1.0f - expf(-dist[t] * ALPHA_C) * w;
    red[threadIdx.x] = term;
    __syncthreads();
#pragma unroll
    for (int s = 128; s > 0; s >>= 1) {
        if (threadIdx.x < s) red[threadIdx.x] += red[threadIdx.x + s];
        __syncthreads();
    }
    if (threadIdx.x == 0) partial[blockIdx.x] = red[0];
}

// =============================================================================
// Kernel 5: fixed-order final reduction -> out[b] = (loss1+loss2)/2
// =============================================================================
__global__ void dac_final_kernel(const float* __restrict__ p1,
                                 const float* __restrict__ p2,
                                 float* __restrict__ out)
{
    const int b = threadIdx.x;
    constexpr int BLK_PER_B = NPTS / 256;                  // 32
    if (b < BATCH) {
        float s1 = 0.0f, s2 = 0.0f;
        for (int k = 0; k < BLK_PER_B; ++k) {
            s1 += p1[b * BLK_PER_B + k];
            s2 += p2[b * BLK_PER_B + k];
        }
        out[b] = (s1 / (float)NPTS + s2 / (float)NPTS) * 0.5f;
    }
}

// =============================================================================
// launcher
// =============================================================================
extern "C" void kernel_launch(void* const* d_in, const int* in_sizes, int n_in,
                              void* d_out, int out_size, void* d_ws, size_t ws_size,
                              hipStream_t stream)
{
    (void)in_sizes; (void)n_in; (void)out_size; (void)ws_size;
    const float* gts   = (const float*)d_in[0];   // [B,N,3]
    const float* preds = (const float*)d_in[1];   // [B,N,3]
    float* out = (float*)d_out;                   // [B]

    // ---- workspace layout (bytes) ----
    char* base = (char*)d_ws;
    const size_t BN = (size_t)BATCH * NPTS;
    int*          idx1   = (int*)          (base + 0 * BN * 4);
    int*          idx2   = (int*)          (base + 1 * BN * 4);
    float*        dist1  = (float*)        (base + 2 * BN * 4);
    float*        dist2  = (float*)        (base + 3 * BN * 4);
    unsigned int* count1 = (unsigned int*) (base + 4 * BN * 4);
    unsigned int* count2 = (unsigned int*) (base + 5 * BN * 4);
    float*        part1  = (float*)        (base + 6 * BN * 4);
    float*        part2  = (float*)        (base + 6 * BN * 4 + (NPTS / 256) * BATCH * 4);

    const int argminGrid = BATCH * (NPTS / ROWS_PER_WG);     // 128
    const int ptGrid     = (BATCH * NPTS + 255) / 256;       // 128
    const int zeroGrid   = (int)((2 * BN + 255) / 256);      // 256

    // zero both count buffers (contiguous)
    dac_zero_kernel<<<zeroGrid, 256, 0, stream>>>(count1, (int)(2 * BN));

    // direction 1: each gt point -> NN in preds ; direction 2: reverse
    dac_argmin_kernel<<<argminGrid, THREADS, 0, stream>>>(gts,   preds, idx1, dist1);
    dac_argmin_kernel<<<argminGrid, THREADS, 0, stream>>>(preds, gts,   idx2, dist2);

    dac_count_kernel<<<ptGrid, 256, 0, stream>>>(idx1, count1);
    dac_count_kernel<<<ptGrid, 256, 0, stream>>>(idx2, count2);

    dac_loss_partial_kernel<<<ptGrid, 256, 0, stream>>>(idx1, dist1, count1, part1);
    dac_loss_partial_kernel<<<ptGrid, 256, 0, stream>>>(idx2, dist2, count2, part2);

    dac_final_kernel<<<1, 32, 0, stream>>>(part1, part2, out);
}